// CorrelationGNN_38130719653939
// MI455X (gfx1250) — compile-verified
//
#include <hip/hip_runtime.h>
#include <hip/hip_bf16.h>

typedef float v2f __attribute__((ext_vector_type(2)));
typedef float v8f __attribute__((ext_vector_type(8)));

#define F_IN 128
#define HID  64

// ---------------- degree / normalization ----------------
__global__ void k_deg_init(float* __restrict__ deg, int N) {
  int n = blockIdx.x * blockDim.x + threadIdx.x;
  if (n < N) deg[n] = 1.0f;  // self-loop counts as 1
}

__global__ void k_deg_count(const long long* __restrict__ dst,
                            float* __restrict__ deg, int E) {
  int e = blockIdx.x * blockDim.x + threadIdx.x;
  if (e < E) atomicAdd(&deg[(int)dst[e]], 1.0f);
}

__global__ void k_dinv(float* __restrict__ degdinv, int N) {
  int n = blockIdx.x * blockDim.x + threadIdx.x;
  if (n < N) degdinv[n] = rsqrtf(degdinv[n]);  // deg >= 1 always
}

// ---------------- layer-1 GEMM: Hs = dinv ⊙ (X @ W1), T1 seeded = Hs ----------------
// 256 threads = 8 waves; each wave computes a 16-row x 64-col output tile with
// V_WMMA_F32_16X16X4_F32 (fp32-exact).
// W1 staged in LDS k-quad interleaved: Wlds[(k/4)*256 + col*4 + (k&3)], so each
// lane's B fragment (W1[k+kh][col], W1[k+kh+1][col]) is one aligned ds_load_b64,
// and a wave's 32 b64 reads cover all 64 LDS banks exactly once (conflict-free).
__global__ __launch_bounds__(256) void k_gemm_wmma(
    const float* __restrict__ X, const float* __restrict__ W1,
    const float* __restrict__ dinv, float* __restrict__ Hs,
    float* __restrict__ T1, int N) {
  __shared__ float Wlds[F_IN * HID];
  for (int i = threadIdx.x; i < F_IN * HID; i += 256) {
    int k = i >> 6, col = i & 63;
    Wlds[(k >> 2) * 256 + col * 4 + (k & 3)] = W1[i];
  }
  __syncthreads();

  const int lane = threadIdx.x & 31;
  const int wave = threadIdx.x >> 5;
  const int rowBase = blockIdx.x * 128 + wave * 16;
  if (rowBase >= N) return;  // wave-uniform: EXEC stays all-ones for WMMA

  const int ml = lane & 15;        // row (A) / col (B,C) within tile
  const int kh = (lane >> 4) << 1; // K sub-offset: 0 (lanes 0-15) or 2 (lanes 16-31)
  int arow = rowBase + ml;
  if (arow >= N) arow = N - 1;     // clamp reads; stores are guarded
  const float* xr = X + (size_t)arow * F_IN;

  v8f acc[4] = {};                 // 4 N-tiles of 16 -> 64 output columns
#pragma unroll 4
  for (int k = 0; k < F_IN; k += 4) {
    // A fragment: 16x4 f32 -> lane holds X[row][k+kh], X[row][k+kh+1]
    v2f a = *(const v2f*)(xr + k + kh);
    const float* wq = Wlds + (k >> 2) * 256 + ml * 4 + kh;
#pragma unroll
    for (int t = 0; t < 4; ++t) {
      // B fragment: one contiguous aligned b64 per lane
      v2f b = *(const v2f*)(wq + t * 64);
      acc[t] = __builtin_amdgcn_wmma_f32_16x16x4_f32(
          false, a, false, b, (short)0, acc[t], false, false);
    }
  }

  // C/D layout: VGPR r -> M = r (lanes 0-15) / M = r+8 (lanes 16-31), N = lane&15
  const int rhalf = (lane >> 4) << 3;
#pragma unroll
  for (int r = 0; r < 8; ++r) {
    int orow = rowBase + r + rhalf;
    if (orow < N) {
      float s = dinv[orow];
      size_t base = (size_t)orow * HID + ml;
#pragma unroll
      for (int t = 0; t < 4; ++t) {
        float v = acc[t][r] * s;
        Hs[base + t * 16] = v;  // pre-scaled rows for the scatter
        T1[base + t * 16] = v;  // self-loop folded into accumulator init
      }
    }
  }
}

// ---------------- layer-1 edge scatter: T1[dst] += Hs[src] ----------------
// One wave per edge; each lane handles 2 consecutive columns (float2 gather,
// 2x global_atomic_add_f32). Hs/T1 are 25.6 MB each -> L2 resident.
// Edge id forced uniform via readfirstlane so src/dst become scalar loads.
__global__ __launch_bounds__(256) void k_scatter1(
    const long long* __restrict__ src, const long long* __restrict__ dst,
    const float* __restrict__ Hs, float* __restrict__ T1, int E) {
  unsigned wid = (blockIdx.x << 3) | (threadIdx.x >> 5);  // 8 waves/block
  unsigned e = (unsigned)__builtin_amdgcn_readfirstlane(wid);
  if (e >= (unsigned)E) return;
  int s = (int)src[e];   // uniform address -> SMEM load
  int d = (int)dst[e];
  int c = (threadIdx.x & 31) << 1;
  v2f h = *(const v2f*)(Hs + (size_t)s * HID + c);
  float* tp = T1 + (size_t)d * HID + c;
  atomicAdd(tp, h.x);
  atomicAdd(tp + 1, h.y);
}

// ---------------- layer-2 prep: H1 = relu(dinv*T1 + b1); Ss = dinv * (H1 . W2) ----------------
__global__ __launch_bounds__(256) void k_layer2(
    const float* __restrict__ T1, const float* __restrict__ b1,
    const float* __restrict__ W2, const float* __restrict__ dinv,
    float* __restrict__ Ss, float* __restrict__ T2, int N) {
  int lane = threadIdx.x & 31;
  int node = blockIdx.x * 8 + (threadIdx.x >> 5);
  if (node >= N) return;
  float dv = dinv[node];
  const float* row = T1 + (size_t)node * HID;
  float h0 = fmaxf(dv * row[lane] + b1[lane], 0.0f);
  float h1 = fmaxf(dv * row[lane + 32] + b1[lane + 32], 0.0f);
  float p = h0 * W2[lane] + h1 * W2[lane + 32];
#pragma unroll
  for (int off = 16; off >= 1; off >>= 1) p += __shfl_xor(p, off, 32);
  if (lane == 0) {
    float v = dv * p;  // fold dinv[src] into the per-node scalar
    Ss[node] = v;
    T2[node] = v;      // self-loop seed
  }
}

// ---------------- layer-2 edge scatter (scalar) ----------------
__global__ void k_scatter2(const long long* __restrict__ src,
                           const long long* __restrict__ dst,
                           const float* __restrict__ Ss,
                           float* __restrict__ T2, int E) {
  int e = blockIdx.x * blockDim.x + threadIdx.x;
  if (e < E) atomicAdd(&T2[(int)dst[e]], Ss[(int)src[e]]);
}

// ---------------- finalize: out = dinv * T2 + b2 ----------------
__global__ void k_final(const float* __restrict__ T2, const float* __restrict__ dinv,
                        const float* __restrict__ b2, float* __restrict__ out, int N) {
  int n = blockIdx.x * blockDim.x + threadIdx.x;
  if (n < N) out[n] = dinv[n] * T2[n] + b2[0];
}

extern "C" void kernel_launch(void* const* d_in, const int* in_sizes, int n_in,
                              void* d_out, int out_size, void* d_ws, size_t ws_size,
                              hipStream_t stream) {
  const float* X       = (const float*)d_in[0];
  const long long* EI  = (const long long*)d_in[1];
  const float* W1      = (const float*)d_in[2];
  const float* b1      = (const float*)d_in[3];
  const float* W2      = (const float*)d_in[4];
  const float* b2      = (const float*)d_in[5];
  const int N = in_sizes[0] / F_IN;
  const int E = in_sizes[1] / 2;
  const long long* src = EI;
  const long long* dst = EI + E;

  // workspace layout (floats): dinv[N] | Hs[64N] | T1[64N] | Ss[N] | T2[N]  (~52.4 MB)
  float* ws   = (float*)d_ws;
  float* dinv = ws;
  float* Hs   = dinv + (size_t)N;
  float* T1   = Hs + (size_t)N * HID;
  float* Ss   = T1 + (size_t)N * HID;
  float* T2   = Ss + (size_t)N;
  float* out  = (float*)d_out;

  const int t = 256;
  k_deg_init <<<(N + t - 1) / t, t, 0, stream>>>(dinv, N);
  k_deg_count<<<(E + t - 1) / t, t, 0, stream>>>(dst, dinv, E);
  k_dinv     <<<(N + t - 1) / t, t, 0, stream>>>(dinv, N);
  k_gemm_wmma<<<(N + 127) / 128, 256, 0, stream>>>(X, W1, dinv, Hs, T1, N);
  k_scatter1 <<<(E + 7) / 8, 256, 0, stream>>>(src, dst, Hs, T1, E);  // 8 edges/block
  k_layer2   <<<(N + 7) / 8, 256, 0, stream>>>(T1, b1, W2, dinv, Ss, T2, N);
  k_scatter2 <<<(E + t - 1) / t, t, 0, stream>>>(src, dst, Ss, T2, E);
  k_final    <<<(N + t - 1) / t, t, 0, stream>>>(T2, dinv, b2, out, N);
}